// DualHOILoss_34789235098114
// MI455X (gfx1250) — compile-verified
//
#include <hip/hip_runtime.h>
#include <hip/hip_bf16.h>
#include <stdint.h>

// Problem constants (from reference): B=64, P=32768, A=32, float32.
#define B_SZ        64
#define P_SZ        32768
#define A_SZ        32
#define N_TOT       (B_SZ * P_SZ)          // 2,097,152 points
#define ANCH_FLOATS (B_SZ * A_SZ * 3)      // 6144 floats = 24 KB
#define TPB1        256
#define GRID1       1024
#define NTHREADS    (TPB1 * GRID1)         // 262,144 threads
#define ITERS       (N_TOT / NTHREADS)     // 8 points per thread
#define TPB2        1024                   // == GRID1, reduces all partials

typedef unsigned int u32x4 __attribute__((ext_vector_type(4)));
typedef int          i32x8 __attribute__((ext_vector_type(8)));
typedef int          i32x4 __attribute__((ext_vector_type(4)));

__global__ __launch_bounds__(TPB1) void hoiloss_stage1(
    const float* __restrict__ anchors,   // (B, A, 3)
    const float* __restrict__ choir,     // (B, P, 3)
    const float* __restrict__ bps,       // (B, P, 3)
    float* __restrict__ partial)         // (GRID1,)
{
    __shared__ float sA[ANCH_FLOATS];    // whole anchor table, 24 KB
    __shared__ float sred[TPB1];

#if __has_builtin(__builtin_amdgcn_tensor_load_to_lds)
    // --- CDNA5 Tensor Data Mover: DMA 24 KB anchors -> LDS (one wave issues).
    // D# built per CDNA5 ISA (1-D tile, data_size=4B).
    if (threadIdx.x < 32u) {             // wave-uniform: only wave 0
        uint64_t ga  = (uint64_t)(uintptr_t)anchors;
        uint32_t lds = (uint32_t)(uintptr_t)(&sA[0]); // LDS offset = low 32 bits
        u32x4 g0;
        g0[0] = 1u;                                          // count=1 (valid), user mode
        g0[1] = lds;                                         // lds_addr (bytes)
        g0[2] = (uint32_t)(ga & 0xFFFFFFFFu);                // global_addr[31:0]
        g0[3] = (uint32_t)((ga >> 32) & 0x1FFFFFFu)          // global_addr[56:32]
              | (2u << 30);                                  // type=2 ("image")
        i32x8 g1;
        g1[0] = (int)(2u << 16);                             // data_size=2 -> 4 bytes; no mask/pad/iterate
        g1[1] = (int)((ANCH_FLOATS & 0xFFFF) << 16);         // tensor_dim0[15:0] in bits[63:48]
        g1[2] = (int)(((ANCH_FLOATS >> 16) & 0xFFFF)         // tensor_dim0[31:16]
              | (1u << 16));                                 // tensor_dim1 = 1
        g1[3] = (int)((ANCH_FLOATS & 0xFFFF) << 16);         // tile_dim0 = 6144 in bits[127:112]
        g1[4] = 1;                                           // tile_dim1 = 1, tile_dim2 = 0
        g1[5] = ANCH_FLOATS;                                 // tensor_dim0_stride[31:0]
        g1[6] = 0;                                           // stride hi / dim1_stride lo
        g1[7] = 0;
        i32x4 gz4 = {0, 0, 0, 0};                            // groups 2/3 unused (<=2D)
        i32x8 gz8 = {0, 0, 0, 0, 0, 0, 0, 0};                // trailing group (clang-23 6-arg form)
        __builtin_amdgcn_tensor_load_to_lds(g0, g1, gz4, gz4, gz8, 0);
        __builtin_amdgcn_s_wait_tensorcnt(0);                // TENSORcnt == 0
    }
#else
    for (int j = (int)threadIdx.x; j < ANCH_FLOATS; j += TPB1) sA[j] = anchors[j];
#endif
    __syncthreads();

    // --- Bandwidth-bound streaming loop. stride = NTHREADS (mult. of 32 and P),
    // so each thread keeps a fixed p (and anchor id = p % 32) and walks b.
    float acc = 0.0f;
    const int tid = (int)(blockIdx.x * TPB1 + threadIdx.x);
    #pragma unroll
    for (int it = 0; it < ITERS; ++it) {
        const int i    = tid + it * NTHREADS;   // flat (b,p) index
        const int p    = i & (P_SZ - 1);
        const int b    = i >> 15;               // P_SZ == 2^15
        const int base = i * 3;
        const float x  = bps[base + 0];
        const float y  = bps[base + 1];
        const float z  = bps[base + 2];
        const float cz = choir[base + 2];       // choir[..., -1]
        const int aoff = (b * A_SZ + (p & (A_SZ - 1))) * 3;
        const float dx = x - sA[aoff + 0];
        const float dy = y - sA[aoff + 1];
        const float dz = z - sA[aoff + 2];
        const float d2 = fmaf(dx, dx, fmaf(dy, dy, dz * dz));
#if __has_builtin(__builtin_amdgcn_sqrtf)
        const float dist = __builtin_amdgcn_sqrtf(d2);
#else
        const float dist = sqrtf(d2);
#endif
        const float e = dist - cz;
        acc = fmaf(e, e, acc);
    }

    // --- Deterministic block reduction (fixed tree order, no atomics).
    sred[threadIdx.x] = acc;
    __syncthreads();
    for (int s = TPB1 / 2; s > 0; s >>= 1) {
        if ((int)threadIdx.x < s) sred[threadIdx.x] += sred[threadIdx.x + s];
        __syncthreads();
    }
    if (threadIdx.x == 0) partial[blockIdx.x] = sred[0];
}

__global__ __launch_bounds__(TPB2) void hoiloss_stage2(
    const float* __restrict__ partial, float* __restrict__ out)
{
    __shared__ float sred[TPB2];
    sred[threadIdx.x] = partial[threadIdx.x];   // GRID1 == TPB2 partials
    __syncthreads();
    for (int s = TPB2 / 2; s > 0; s >>= 1) {
        if ((int)threadIdx.x < s) sred[threadIdx.x] += sred[threadIdx.x + s];
        __syncthreads();
    }
    if (threadIdx.x == 0) {
        out[0] = sred[0] * (1.0f / (float)N_TOT);  // choir_loss (mean)
        out[1] = 0.0f;                             // hand_contact_loss
    }
}

extern "C" void kernel_launch(void* const* d_in, const int* in_sizes, int n_in,
                              void* d_out, int out_size, void* d_ws, size_t ws_size,
                              hipStream_t stream) {
    (void)in_sizes; (void)n_in; (void)out_size; (void)ws_size;
    const float* anchors = (const float*)d_in[0];   // (64, 32, 3)
    const float* choir   = (const float*)d_in[1];   // (64, 32768, 3)
    const float* bps     = (const float*)d_in[2];   // (64, 32768, 3)
    float* out     = (float*)d_out;                 // 2 floats
    float* partial = (float*)d_ws;                  // 4 KB scratch

    hoiloss_stage1<<<GRID1, TPB1, 0, stream>>>(anchors, choir, bps, partial);
    hoiloss_stage2<<<1, TPB2, 0, stream>>>(partial, out);
}